// SO2TFN_38903813767518
// MI455X (gfx1250) — compile-verified
//
#include <hip/hip_runtime.h>
#include <hip/hip_bf16.h>
#include <math.h>

typedef float v2f __attribute__((ext_vector_type(2)));
typedef float v8f __attribute__((ext_vector_type(8)));

#define NNODE 50000
#define NEDGE 800000
#define NGRAPH 64
#define HCH 32

// ---------------------------------------------------------------- helpers

__device__ __forceinline__ float gaussv(float d, int k) {
  // mus = linspace(0.7, 1.7, 10); sigma = 0.1 -> exp(-50*(d-mu)^2)
  if (k >= 10) return 0.0f;
  float mu = 0.7f + (float)k * (1.0f / 9.0f);
  float t = d - mu;
  return __expf(-50.0f * t * t);
}

__device__ __forceinline__ float cutoffv(float d) {
  float u = d * (1.0f / 1.5f);
  float u2 = u * u, u3 = u2 * u;
  float u6 = u3 * u3, u7 = u6 * u, u8 = u7 * u;
  float f = 1.0f - 28.0f * u6 + 48.0f * u7 - 21.0f * u8;
  return (u < 1.0f) ? f : 0.0f;
}

// Cooperative load of W (NB=10 x C) into LDS, zero-padded to K=12 rows.
template <int C>
__device__ __forceinline__ void load_W_lds(int tid, const float* __restrict__ W,
                                           float* __restrict__ Wsh) {
  for (int idx = tid; idx < 12 * C; idx += 128) {
    int kk = idx / C;
    int cc = idx - kk * C;
    Wsh[idx] = (kk < 10) ? W[kk * C + cc] : 0.0f;
  }
}

// Per-wave WMMA GEMM: w[16 edges][C] = emb(16x10, K padded to 12) @ W(10xC) + b
// A fragment (16x4 f32): lanes 0-15 hold K={0,1}, lanes 16-31 hold K={2,3}.
// B fragment (4x16 f32): lanes 0-15 hold K={0,1}, lanes 16-31 hold K={2,3}, N=lane&15.
// C/D fragment: VGPR r, lane l -> M = r + 8*(l>=16), N = l&15.
template <int C>
__device__ __forceinline__ void compute_edge_weights(int lane, float d,
                                                     const float* __restrict__ Wsh,
                                                     const float* __restrict__ b,
                                                     float* __restrict__ wsrow) {
  const int row = lane & 15;
  const int khalf = lane >> 4;
  v2f a[3];
#pragma unroll
  for (int s = 0; s < 3; ++s) {
    int k0 = s * 4 + khalf * 2;
    a[s].x = gaussv(d, k0);
    a[s].y = gaussv(d, k0 + 1);
  }
#pragma unroll
  for (int t = 0; t < C / 16; ++t) {
    v8f acc;
    float bias = b[t * 16 + row];
#pragma unroll
    for (int r = 0; r < 8; ++r) acc[r] = bias;
#pragma unroll
    for (int s = 0; s < 3; ++s) {
      int k0 = s * 4 + khalf * 2;
      v2f bf;
      bf.x = Wsh[k0 * C + t * 16 + row];        // unconditional ds_load (padded)
      bf.y = Wsh[(k0 + 1) * C + t * 16 + row];
      acc = __builtin_amdgcn_wmma_f32_16x16x4_f32(false, a[s], false, bf,
                                                  (short)0, acc, false, false);
    }
#pragma unroll
    for (int r = 0; r < 8; ++r) {
      int M = r + khalf * 8;
      wsrow[M * C + t * 16 + row] = acc[r];
    }
  }
}

// ---------------------------------------------------------------- layer 1
__global__ void __launch_bounds__(128) edge_layer1_kernel(
    const int* __restrict__ ei, const float* __restrict__ elen,
    const float* __restrict__ D1, const float* __restrict__ x,
    const float* __restrict__ W1, const float* __restrict__ b1,
    float* __restrict__ hacc, int E) {
  __shared__ float ws[4][16 * 64];
  __shared__ float Wsh[12 * 64];
  const int tid = threadIdx.x;
  const int wave = tid >> 5, lane = tid & 31;
  const int eb = blockIdx.x * 64 + wave * 16;

  load_W_lds<64>(tid, W1, Wsh);

  int eidx = eb + (lane & 15);
  if (eidx >= E) eidx = E - 1;
  int srcv = ei[eidx];
  int dstv = ei[E + eidx];
  float dv = elen[eidx];
  float xv = x[srcv];
  __builtin_prefetch(D1 + (size_t)eb * 16 + lane * 8, 0, 3);  // 1KB D tile

  __syncthreads();
  compute_edge_weights<64>(lane, dv, Wsh, b1, ws[wave]);

  const int c = lane;
  for (int i = 0; i < 16; ++i) {
    int e = eb + i;
    if (e >= E) break;
    int dst = __shfl(dstv, i);
    float d = __shfl(dv, i);
    float xs = __shfl(xv, i);
    float ew = cutoffv(d);
    const float4* D4 = (const float4*)(D1 + (size_t)e * 16);
    float4 dr0 = D4[0];  // D1[0][*]
    float4 dr2 = D4[2];  // D1[2][*]
    float xj0 = dr0.x * xs;  // D0 = D1[0][0]
    const float* wr = ws[wave] + i * 64;
    float o0 = xj0 * wr[c];
    float o2 = xj0 * wr[32 + c];
    float* ho = hacc + (size_t)dst * 128;
    float d0[4] = {dr0.x, dr0.y, dr0.z, dr0.w};
    float d2[4] = {dr2.x, dr2.y, dr2.z, dr2.w};
#pragma unroll
    for (int r = 0; r < 4; ++r) {
      float m = ew * (d0[r] * o0 + d2[r] * o2);  // DT1[r][n] = D1[n][r]
      atomicAdd(&ho[r * 32 + c], m);
    }
  }
}

// ---------------------------------------------------------------- layer 2
__global__ void __launch_bounds__(128) edge_layer2_kernel(
    const int* __restrict__ ei, const float* __restrict__ elen,
    const float* __restrict__ D1, const float* __restrict__ hA,
    const float* __restrict__ W2, const float* __restrict__ b2,
    float* __restrict__ hacc, int E) {
  __shared__ float ws[4][16 * 192];
  __shared__ float Wsh[12 * 192];
  const int tid = threadIdx.x;
  const int wave = tid >> 5, lane = tid & 31;
  const int eb = blockIdx.x * 64 + wave * 16;

  load_W_lds<192>(tid, W2, Wsh);

  int eidx = eb + (lane & 15);
  if (eidx >= E) eidx = E - 1;
  int srcv = ei[eidx];
  int dstv = ei[E + eidx];
  float dv = elen[eidx];
  // prefetch: lane l and l+16 share srcv -> together cover the full 512B row
  __builtin_prefetch(hA + (size_t)srcv * 128 + (lane >> 4) * 64, 0, 3);
  __builtin_prefetch(D1 + (size_t)eb * 16 + lane * 8, 0, 3);

  __syncthreads();
  compute_edge_weights<192>(lane, dv, Wsh, b2, ws[wave]);

  const int c = lane;
  for (int i = 0; i < 16; ++i) {
    int e = eb + i;
    if (e >= E) break;
    int src = __shfl(srcv, i);
    int dst = __shfl(dstv, i);
    float d = __shfl(dv, i);
    float ew = cutoffv(d);
    const float4* D4 = (const float4*)(D1 + (size_t)e * 16);
    float4 dr0 = D4[0], dr1 = D4[1], dr2 = D4[2], dr3 = D4[3];
    const float* hs = hA + (size_t)src * 128;
    float ha0 = hs[c], ha1 = hs[32 + c], ha2 = hs[64 + c], ha3 = hs[96 + c];
    float xj0 = dr0.x * ha0 + dr0.y * ha1 + dr0.z * ha2 + dr0.w * ha3;
    float xj1 = dr1.x * ha0 + dr1.y * ha1 + dr1.z * ha2 + dr1.w * ha3;
    float xj2 = dr2.x * ha0 + dr2.y * ha1 + dr2.z * ha2 + dr2.w * ha3;
    float xj3 = dr3.x * ha0 + dr3.y * ha1 + dr3.z * ha2 + dr3.w * ha3;
    const float* wr = ws[wave] + i * 192;
    float w0 = wr[c], w1 = wr[32 + c], w2 = wr[64 + c];
    float w3 = wr[96 + c], w4 = wr[128 + c], w5 = wr[160 + c];
    float o0 = w0 * xj0 + w2 * xj2;
    float o2 = w1 * xj0 + w3 * xj2;
    float o3 = w4 * xj3 - w5 * xj1;
    float o1 = w5 * xj3 + w4 * xj1;
    float* ho = hacc + (size_t)dst * 128;
    float d0[4] = {dr0.x, dr0.y, dr0.z, dr0.w};
    float d1[4] = {dr1.x, dr1.y, dr1.z, dr1.w};
    float d2[4] = {dr2.x, dr2.y, dr2.z, dr2.w};
    float d3[4] = {dr3.x, dr3.y, dr3.z, dr3.w};
#pragma unroll
    for (int r = 0; r < 4; ++r) {
      float m = ew * (d0[r] * o0 + d1[r] * o1 + d2[r] * o2 + d3[r] * o3);
      atomicAdd(&ho[r * 32 + c], m);
    }
  }
}

// ---------------------------------------------------------------- layer 3
__global__ void __launch_bounds__(128) edge_layer3_kernel(
    const int* __restrict__ ei, const float* __restrict__ elen,
    const float* __restrict__ D1, const float* __restrict__ hB,
    const float* __restrict__ W3, const float* __restrict__ b3,
    float* __restrict__ h3out, int E) {
  __shared__ float ws[4][16 * 64];
  __shared__ float Wsh[12 * 64];
  const int tid = threadIdx.x;
  const int wave = tid >> 5, lane = tid & 31;
  const int eb = blockIdx.x * 64 + wave * 16;

  load_W_lds<64>(tid, W3, Wsh);

  int eidx = eb + (lane & 15);
  if (eidx >= E) eidx = E - 1;
  int srcv = ei[eidx];
  int dstv = ei[E + eidx];
  float dv = elen[eidx];
  __builtin_prefetch(hB + (size_t)srcv * 128 + (lane >> 4) * 64, 0, 3);
  __builtin_prefetch(D1 + (size_t)eb * 16 + lane * 8, 0, 3);

  __syncthreads();
  compute_edge_weights<64>(lane, dv, Wsh, b3, ws[wave]);

  const int c = lane;
  for (int i = 0; i < 16; ++i) {
    int e = eb + i;
    if (e >= E) break;
    int src = __shfl(srcv, i);
    int dst = __shfl(dstv, i);
    float d = __shfl(dv, i);
    float ew = cutoffv(d);
    const float4* D4 = (const float4*)(D1 + (size_t)e * 16);
    float4 dr0 = D4[0], dr2 = D4[2];
    const float* hs = hB + (size_t)src * 128;
    float ha0 = hs[c], ha1 = hs[32 + c], ha2 = hs[64 + c], ha3 = hs[96 + c];
    float xj0 = dr0.x * ha0 + dr0.y * ha1 + dr0.z * ha2 + dr0.w * ha3;
    float xj2 = dr2.x * ha0 + dr2.y * ha1 + dr2.z * ha2 + dr2.w * ha3;
    const float* wr = ws[wave] + i * 64;
    float o0 = wr[c] * xj0 + wr[32 + c] * xj2;
    float m = ew * dr0.x * o0;  // DT0 = D1[0][0]
    atomicAdd(&h3out[(size_t)dst * 32 + c], m);
  }
}

// -------------------------------------------------- node update: dw + norm_act
__global__ void __launch_bounds__(256) node_update_kernel(
    const float* __restrict__ hin, const float* __restrict__ si,
    float* __restrict__ hout, int Nn) {
  int gid = blockIdx.x * blockDim.x + threadIdx.x;
  int node = gid >> 5, c = gid & 31;
  if (node >= Nn) return;
  const float* hn = hin + (size_t)node * 128;
  float a0 = 0.f, a1 = 0.f, a2 = 0.f, a3 = 0.f;
#pragma unroll 8
  for (int k = 0; k < 32; ++k) {
    float w0 = si[k * 32 + c];          // si[0][k][c]
    float w1 = si[1024 + k * 32 + c];   // si[1][k][c]
    a0 += hn[k] * w0;
    a1 += hn[32 + k] * w1;
    a2 += hn[64 + k] * w1;
    a3 += hn[96 + k] * w1;
  }
  float n0 = sqrtf(a0 * a0 + 1e-8f);
  float n1 = sqrtf(a1 * a1 + a2 * a2 + a3 * a3 + 1e-8f);
  const float LN2 = 0.69314718055994531f;
  float s0 = (log1pf(__expf(n0)) - LN2) / n0;
  float s1 = (log1pf(__expf(n1)) - LN2) / n1;
  float* ho = hout + (size_t)node * 128;
  ho[c] = a0 * s0;
  ho[32 + c] = a1 * s1;
  ho[64 + c] = a2 * s1;
  ho[96 + c] = a3 * s1;
}

// -------------------------------------------------- final node: silu + pool
__global__ void __launch_bounds__(256) node_final_kernel(
    const float* __restrict__ h3, const float* __restrict__ si3,
    const int* __restrict__ batch, float* __restrict__ pooled, int Nn) {
  int gid = blockIdx.x * blockDim.x + threadIdx.x;
  int node = gid >> 5, c = gid & 31;
  if (node >= Nn) return;
  const float* hn = h3 + (size_t)node * 32;
  float a = 0.f;
#pragma unroll 8
  for (int k = 0; k < 32; ++k) a += hn[k] * si3[k * 32 + c];
  float s = a / (1.0f + __expf(-a));  // silu
  atomicAdd(&pooled[batch[node] * 32 + c], s);
}

// -------------------------------------------------- readout: pooled@Wo+softmax
__global__ void out_kernel(const float* __restrict__ pooled,
                           const float* __restrict__ Wo,
                           const float* __restrict__ bo,
                           float* __restrict__ out) {
  int g = threadIdx.x;
  if (g >= NGRAPH) return;
  float lg[8];
#pragma unroll
  for (int j = 0; j < 8; ++j) lg[j] = bo[j];
  for (int k = 0; k < 32; ++k) {
    float p = pooled[g * 32 + k];
#pragma unroll
    for (int j = 0; j < 8; ++j) lg[j] += p * Wo[k * 8 + j];
  }
  float mx = lg[0];
#pragma unroll
  for (int j = 1; j < 8; ++j) mx = fmaxf(mx, lg[j]);
  float sum = 0.f;
#pragma unroll
  for (int j = 0; j < 8; ++j) { lg[j] = __expf(lg[j] - mx); sum += lg[j]; }
  float inv = 1.0f / sum;
#pragma unroll
  for (int j = 0; j < 8; ++j) out[g * 8 + j] = lg[j] * inv;
}

// ----------------------------------------------------------------- driver
extern "C" void kernel_launch(void* const* d_in, const int* in_sizes, int n_in,
                              void* d_out, int out_size, void* d_ws, size_t ws_size,
                              hipStream_t stream) {
  const float* x    = (const float*)d_in[0];
  const int*   ei   = (const int*)d_in[1];
  const float* D1   = (const float*)d_in[2];
  const float* elen = (const float*)d_in[3];
  const int*   batch= (const int*)d_in[4];
  const float* W1   = (const float*)d_in[5];
  const float* b1   = (const float*)d_in[6];
  const float* W2   = (const float*)d_in[7];
  const float* b2   = (const float*)d_in[8];
  const float* W3   = (const float*)d_in[9];
  const float* b3   = (const float*)d_in[10];
  const float* si1  = (const float*)d_in[11];
  const float* si2  = (const float*)d_in[12];
  const float* si3  = (const float*)d_in[13];
  const float* Wo   = (const float*)d_in[14];
  const float* bo   = (const float*)d_in[15];
  float* out = (float*)d_out;

  const int N = NNODE, E = NEDGE, G = NGRAPH;
  float* hacc   = (float*)d_ws;                 // N*128
  float* hnode  = hacc + (size_t)N * 128;       // N*128
  float* h3buf  = hnode + (size_t)N * 128;      // N*32
  float* pooled = h3buf + (size_t)N * 32;       // G*32

  const int egrid = (E + 63) / 64;          // 12500
  const int ngrid = (N * 32 + 255) / 256;   // 6250

  hipMemsetAsync(hacc, 0, (size_t)N * 128 * sizeof(float), stream);
  edge_layer1_kernel<<<egrid, 128, 0, stream>>>(ei, elen, D1, x, W1, b1, hacc, E);
  node_update_kernel<<<ngrid, 256, 0, stream>>>(hacc, si1, hnode, N);

  hipMemsetAsync(hacc, 0, (size_t)N * 128 * sizeof(float), stream);
  edge_layer2_kernel<<<egrid, 128, 0, stream>>>(ei, elen, D1, hnode, W2, b2, hacc, E);
  node_update_kernel<<<ngrid, 256, 0, stream>>>(hacc, si2, hnode, N);

  hipMemsetAsync(h3buf, 0, (size_t)N * 32 * sizeof(float), stream);
  edge_layer3_kernel<<<egrid, 128, 0, stream>>>(ei, elen, D1, hnode, W3, b3, h3buf, E);

  hipMemsetAsync(pooled, 0, (size_t)G * 32 * sizeof(float), stream);
  node_final_kernel<<<ngrid, 256, 0, stream>>>(h3buf, si3, batch, pooled, N);
  out_kernel<<<1, 64, 0, stream>>>(pooled, Wo, bo, out);
}